// ScaleSpaceAffinePatchExtractor_39049842655335
// MI455X (gfx1250) — compile-verified
//
#include <hip/hip_runtime.h>
#include <cmath>
#include <algorithm>

// ---------------------------------------------------------------------------
// ScaleSpaceAffinePatchExtractor for MI455X (gfx1250, wave32).
// Bandwidth-bound pipeline (~1GB traffic -> ~45us floor @ 23.3TB/s).
// Separable Gaussian blurs run on V_WMMA_F32_16X16X4_F32 (exact f32 math):
// one wave = one 16x16 output tile; O = A(16xK) x B(Kx16), K = 16+2r.
// Kernels are templated on <radius, dim> so the K-loop fully unrolls,
// band-weight reads are unconditional (pre-padded LDS table), and stores /
// vertical loads use immediate offsets from a single base address.
// ---------------------------------------------------------------------------

typedef float v2f __attribute__((ext_vector_type(2)));
typedef float v8f __attribute__((ext_vector_type(8)));

#if defined(__has_builtin)
#  if __has_builtin(__builtin_amdgcn_wmma_f32_16x16x4_f32)
#    define HAVE_WMMA_F32X4 1
#  endif
#endif
#ifndef HAVE_WMMA_F32X4
#  define HAVE_WMMA_F32X4 0
#endif

#define WPB        4          // waves per block (wave32)
#define MAX_TAPS   24         // r <= 8 -> taps <= 17
#define TSTRIDE    33         // LDS tile row stride (odd -> conflict-free)
#define WPAD       16         // zero-pad before weight band in LDS
#define N_FEATS    4000
#define SEL_CAP    4096
#define CAND_MAX   (1 << 21)

__device__ __forceinline__ int d_min(int a, int b) { return a < b ? a : b; }
__device__ __forceinline__ int d_max(int a, int b) { return a > b ? a : b; }

__device__ __forceinline__ v8f wmma_f32x4(v2f a, v2f b, v8f c) {
#if HAVE_WMMA_F32X4
  return __builtin_amdgcn_wmma_f32_16x16x4_f32(false, a, false, b,
                                               (short)0, c, false, false);
#else
  // layout-identical scalar fallback (never taken on gfx1250)
  return c;
#endif
}

// Build zero-padded, normalized Gaussian band in LDS: sW[WPAD+t] = w[t]/sum,
// zero elsewhere. All in-loop band reads are then unconditional.
template <int TAPS, int R>
__device__ __forceinline__ void build_weights(float* sW, float sigma, int tid) {
  if (tid < 64) {
    int t = tid - WPAD;
    float v = 0.0f;
    if (t >= 0 && t < TAPS) {
      float d = (float)(t - R);
      v = expf(-0.5f * d * d / (sigma * sigma));
    }
    sW[tid] = v;
  }
  __syncthreads();
  float wsum = 0.0f;
#pragma unroll
  for (int t = 0; t < TAPS; ++t) wsum += sW[WPAD + t];
  const float inv = 1.0f / wsum;
  __syncthreads();
  if (tid < 64) sW[tid] *= inv;
  __syncthreads();
}

// ---------------------------------------------------------------------------
// Horizontal blur: A = image tile (LDS), B = banded weights.
// ---------------------------------------------------------------------------
template <int R, int DIM>
__global__ __launch_bounds__(128) void k_blur_h_t(const float* __restrict__ in,
                                                  float* __restrict__ out,
                                                  float sigma) {
  constexpr int TAPS = 2 * R + 1;
  constexpr int KT = (16 + 2 * R + 3) & ~3;
  constexpr int TILESX = DIM >> 4;
  constexpr int LOGT = (DIM == 2048) ? 7 : (DIM == 1024) ? 6 : 5;

  __shared__ float sW[64];
  __shared__ float sTile[WPB][16 * TSTRIDE];
  const int tid = threadIdx.x;
  build_weights<TAPS, R>(sW, sigma, tid);

  const int wave = tid >> 5, lane = tid & 31;
  const int tileId = blockIdx.x * WPB + wave;
  const int tx = tileId & (TILESX - 1);
  const int ty = tileId >> LOGT;
  const int row0 = ty << 4, col0 = tx << 4;

  float* tile = sTile[wave];
#pragma unroll
  for (int it = 0; it < KT / 2; ++it) {          // 16*KT/32 exact
    int idx = lane + 32 * it;
    int rr = idx / KT;                            // constexpr divisor
    int cc = idx - rr * KT;
    int gc = col0 - R + cc;
    int gcc = d_min(d_max(gc, 0), DIM - 1);       // branchless zero-pad
    float v = in[(size_t)(row0 + rr) * DIM + gcc];
    tile[rr * TSTRIDE + cc] = (gc == gcc) ? v : 0.0f;
  }
  __syncthreads();

  const int half = lane >> 4;                     // A: lanes16-31 hold K+2/K+3
  const int m = lane & 15;                        // M for A, N for B/D
  v8f acc = {};
#pragma unroll
  for (int k0 = 0; k0 < KT; k0 += 4) {
    const int j0 = k0 + 2 * half;
    v2f a, b;
    a.x = tile[m * TSTRIDE + j0];
    a.y = tile[m * TSTRIDE + j0 + 1];
    b.x = sW[WPAD + j0 - m];                      // unconditional band read
    b.y = sW[WPAD + j0 - m + 1];
    acc = wmma_f32x4(a, b, acc);
  }
  float* ob = out + (size_t)(row0 + 8 * half) * DIM + col0 + m;
#pragma unroll
  for (int p = 0; p < 8; ++p) ob[(size_t)p * DIM] = acc[p];  // imm offsets
}

// ---------------------------------------------------------------------------
// Vertical blur: A = banded weights, B = image rows (coalesced 64B loads).
// Wave-uniform interior fast path avoids all bounds handling.
// ---------------------------------------------------------------------------
template <int R, int DIM>
__global__ __launch_bounds__(128) void k_blur_v_t(const float* __restrict__ in,
                                                  float* __restrict__ out,
                                                  float sigma) {
  constexpr int TAPS = 2 * R + 1;
  constexpr int KT = (16 + 2 * R + 3) & ~3;
  constexpr int TILESX = DIM >> 4;
  constexpr int LOGT = (DIM == 2048) ? 7 : (DIM == 1024) ? 6 : 5;

  __shared__ float sW[64];
  const int tid = threadIdx.x;
  build_weights<TAPS, R>(sW, sigma, tid);

  const int wave = tid >> 5, lane = tid & 31;
  const int tileId = blockIdx.x * WPB + wave;
  const int tx = tileId & (TILESX - 1);
  const int ty = tileId >> LOGT;
  const int row0 = ty << 4, col0 = tx << 4;
  const int half = lane >> 4;
  const int m = lane & 15;

  v8f acc = {};
  if (row0 >= R && row0 <= DIM - (KT - R)) {
    // interior: unconditional loads, single base + immediate row offsets
    const float* p0 = in + (size_t)(row0 - R + 2 * half) * DIM + col0 + m;
#pragma unroll
    for (int k0 = 0; k0 < KT; k0 += 4) {
      const int j0 = k0 + 2 * half;
      v2f a, b;
      a.x = sW[WPAD + j0 - m];
      a.y = sW[WPAD + j0 - m + 1];
      b.x = p0[(size_t)k0 * DIM];
      b.y = p0[(size_t)(k0 + 1) * DIM];
      acc = wmma_f32x4(a, b, acc);
    }
  } else {
    // boundary: branchless clamp + select (zero padding)
#pragma unroll
    for (int k0 = 0; k0 < KT; k0 += 4) {
      const int j0 = k0 + 2 * half;
      v2f a, b;
      a.x = sW[WPAD + j0 - m];
      a.y = sW[WPAD + j0 - m + 1];
      int rb = row0 - R + j0;
      int r0c = d_min(d_max(rb, 0), DIM - 1);
      int r1c = d_min(d_max(rb + 1, 0), DIM - 1);
      float v0 = in[(size_t)r0c * DIM + col0 + m];
      float v1 = in[(size_t)r1c * DIM + col0 + m];
      b.x = (rb == r0c) ? v0 : 0.0f;
      b.y = (rb + 1 == r1c) ? v1 : 0.0f;
      acc = wmma_f32x4(a, b, acc);
    }
  }
  float* ob = out + (size_t)(row0 + 8 * half) * DIM + col0 + m;
#pragma unroll
  for (int p = 0; p < 8; ++p) ob[(size_t)p * DIM] = acc[p];
}

// ---------------------------------------------------------------------------
// Generic runtime-radius fallback blurs (dispatcher default branch only)
// ---------------------------------------------------------------------------
__global__ __launch_bounds__(128) void k_blur_h_gen(const float* __restrict__ in,
                                                    float* __restrict__ out,
                                                    int W, float sigma, int r) {
  __shared__ float sW[64];
  __shared__ float sTile[WPB][16 * TSTRIDE];
  const int taps = 2 * r + 1;
  const int tid = threadIdx.x;
  if (tid < 64) {
    int t = tid - WPAD;
    float v = 0.0f;
    if (t >= 0 && t < taps) {
      float d = (float)(t - r);
      v = expf(-0.5f * d * d / (sigma * sigma));
    }
    sW[tid] = v;
  }
  __syncthreads();
  float wsum = 0.0f;
  for (int t = 0; t < taps; ++t) wsum += sW[WPAD + t];
  const float inv = 1.0f / wsum;
  __syncthreads();
  if (tid < 64) sW[tid] *= inv;
  __syncthreads();

  const int wave = tid >> 5, lane = tid & 31;
  const int tilesX = W >> 4;
  const int tileId = blockIdx.x * WPB + wave;
  const int tx = tileId % tilesX, ty = tileId / tilesX;
  const int row0 = ty << 4, col0 = tx << 4;
  const int Kt = (16 + 2 * r + 3) & ~3;

  float* tile = sTile[wave];
  for (int idx = lane; idx < 16 * Kt; idx += 32) {
    int rr = idx / Kt, cc = idx - rr * Kt;
    int gc = col0 - r + cc;
    int gcc = d_min(d_max(gc, 0), W - 1);
    float v = in[(size_t)(row0 + rr) * W + gcc];
    tile[rr * TSTRIDE + cc] = (gc == gcc) ? v : 0.0f;
  }
  __syncthreads();
  const int half = lane >> 4, m = lane & 15;
  v8f acc = {};
  for (int k0 = 0; k0 < Kt; k0 += 4) {
    const int j0 = k0 + 2 * half;
    v2f a, b;
    a.x = tile[m * TSTRIDE + j0];
    a.y = tile[m * TSTRIDE + j0 + 1];
    b.x = sW[WPAD + j0 - m];
    b.y = sW[WPAD + j0 - m + 1];
    acc = wmma_f32x4(a, b, acc);
  }
  for (int p = 0; p < 8; ++p)
    out[(size_t)(row0 + p + 8 * half) * W + col0 + m] = acc[p];
}

__global__ __launch_bounds__(128) void k_blur_v_gen(const float* __restrict__ in,
                                                    float* __restrict__ out,
                                                    int H, int W, float sigma, int r) {
  __shared__ float sW[64];
  const int taps = 2 * r + 1;
  const int tid = threadIdx.x;
  if (tid < 64) {
    int t = tid - WPAD;
    float v = 0.0f;
    if (t >= 0 && t < taps) {
      float d = (float)(t - r);
      v = expf(-0.5f * d * d / (sigma * sigma));
    }
    sW[tid] = v;
  }
  __syncthreads();
  float wsum = 0.0f;
  for (int t = 0; t < taps; ++t) wsum += sW[WPAD + t];
  const float inv = 1.0f / wsum;
  __syncthreads();
  if (tid < 64) sW[tid] *= inv;
  __syncthreads();

  const int wave = tid >> 5, lane = tid & 31;
  const int tilesX = W >> 4;
  const int tileId = blockIdx.x * WPB + wave;
  const int tx = tileId % tilesX, ty = tileId / tilesX;
  const int row0 = ty << 4, col0 = tx << 4;
  const int Kt = (16 + 2 * r + 3) & ~3;
  const int half = lane >> 4, m = lane & 15;

  v8f acc = {};
  for (int k0 = 0; k0 < Kt; k0 += 4) {
    const int j0 = k0 + 2 * half;
    v2f a, b;
    a.x = sW[WPAD + j0 - m];
    a.y = sW[WPAD + j0 - m + 1];
    int rb = row0 - r + j0;
    int r0c = d_min(d_max(rb, 0), H - 1);
    int r1c = d_min(d_max(rb + 1, 0), H - 1);
    float v0 = in[(size_t)r0c * W + col0 + m];
    float v1 = in[(size_t)r1c * W + col0 + m];
    b.x = (rb == r0c) ? v0 : 0.0f;
    b.y = (rb + 1 == r1c) ? v1 : 0.0f;
    acc = wmma_f32x4(a, b, acc);
  }
  for (int p = 0; p < 8; ++p)
    out[(size_t)(row0 + p + 8 * half) * W + col0 + m] = acc[p];
}

// ---------------------------------------------------------------------------
// Hessian-determinant response (edge-padded stencil); W = 1<<lw
// ---------------------------------------------------------------------------
__global__ void k_hessian(const float* __restrict__ in, float* __restrict__ resp,
                          int H, int lw, float s4) {
  const int W = 1 << lw;
  int idx = blockIdx.x * blockDim.x + threadIdx.x;
  if (idx >= (H << lw)) return;
  int y = idx >> lw, x = idx & (W - 1);
  int xm = d_max(x - 1, 0), xp = d_min(x + 1, W - 1);
  int ym = d_max(y - 1, 0), yp = d_min(y + 1, H - 1);
  float c = in[idx];
  float gxx = in[(y << lw) + xp] - 2.0f * c + in[(y << lw) + xm];
  float gyy = in[(yp << lw) + x] - 2.0f * c + in[(ym << lw) + x];
  float gxy = 0.25f * (in[(yp << lw) + xp] - in[(yp << lw) + xm]
                     - in[(ym << lw) + xp] + in[(ym << lw) + xm]);
  resp[idx] = s4 * (gxx * gyy - gxy * gxy);
}

__global__ void k_downsample(const float* __restrict__ in, float* __restrict__ out,
                             int n2, int lw2) {
  int idx = blockIdx.x * blockDim.x + threadIdx.x;
  if (idx >= n2) return;
  int y = idx >> lw2, x = idx & ((1 << lw2) - 1);
  out[idx] = in[((size_t)y << (lw2 + 2)) + 2 * x];
}

// ---------------------------------------------------------------------------
// 3x3x3 scale-space NMS -> append candidates (val, packed oct/lvl/row/col)
// ---------------------------------------------------------------------------
__global__ void k_nms(const float* __restrict__ lo, const float* __restrict__ ce,
                      const float* __restrict__ hi, int H, int lw, int b,
                      int oct, int lvl, float* __restrict__ cv,
                      unsigned* __restrict__ ci, unsigned* __restrict__ cnt,
                      int cmax) {
  const int W = 1 << lw;
  int idx = blockIdx.x * blockDim.x + threadIdx.x;
  if (idx >= (H << lw)) return;
  int y = idx >> lw, x = idx & (W - 1);
  if (y < b || y >= H - b || x < b || x >= W - b) return;
  float c = ce[idx];
  if (!(c > 0.0f)) return;
  float m = -3.402823466e38f;
#pragma unroll
  for (int dy = -1; dy <= 1; ++dy)
#pragma unroll
    for (int dx = -1; dx <= 1; ++dx) {
      int p = ((y + dy) << lw) + (x + dx);
      m = fmaxf(m, fmaxf(lo[p], fmaxf(hi[p], ce[p])));
    }
  if (c >= m) {
    unsigned pos = atomicAdd(cnt, 1u);
    if ((int)pos < cmax) {
      cv[pos] = c;
      ci[pos] = ((unsigned)oct << 24) | ((unsigned)lvl << 22)
              | ((unsigned)y << 11) | (unsigned)x;
    }
  }
}

// ---------------------------------------------------------------------------
// Exact top-N via 4-round radix select on positive-float bit patterns
// state: [0]=prefix  [1]=kRemaining  [2]=threshold T  [3]=#ties to take
// ---------------------------------------------------------------------------
__global__ void k_sel_init(unsigned* state) { state[0] = 0u; state[1] = N_FEATS; }

__global__ void k_hist(const float* __restrict__ cv, const unsigned* __restrict__ cnt,
                       unsigned* __restrict__ hist, const unsigned* __restrict__ state,
                       int rnd, int cmax) {
  int n = d_min((int)*cnt, cmax);
  int i = blockIdx.x * blockDim.x + threadIdx.x;
  if (i >= n) return;
  unsigned bits = __float_as_uint(cv[i]);
  if (rnd > 0 && (bits >> (32 - 8 * rnd)) != state[0]) return;
  atomicAdd(&hist[(bits >> (24 - 8 * rnd)) & 0xFFu], 1u);
}

__global__ void k_scan(const unsigned* __restrict__ hist, unsigned* state, int rnd) {
  unsigned kRem = state[1], prefix = state[0];
  unsigned acc = 0; int sel = 0;
  for (int bin = 255; bin >= 0; --bin) {
    unsigned h = hist[bin];
    if (acc + h >= kRem || bin == 0) { sel = bin; break; }
    acc += h;
  }
  state[0] = (prefix << 8) | (unsigned)sel;
  state[1] = kRem - acc;
  if (rnd == 3) { state[2] = state[0]; state[3] = kRem - acc; }
}

__global__ void k_compact(const float* __restrict__ cv, const unsigned* __restrict__ ci,
                          const unsigned* __restrict__ cnt, const unsigned* __restrict__ state,
                          float* __restrict__ sv, unsigned* __restrict__ si,
                          unsigned* selCnt, unsigned* eqCnt, int cmax) {
  int n = d_min((int)*cnt, cmax);
  int i = blockIdx.x * blockDim.x + threadIdx.x;
  if (i >= n) return;
  unsigned bits = __float_as_uint(cv[i]);
  unsigned T = state[2], eqTake = state[3];
  if (bits > T) {
    unsigned p = atomicAdd(selCnt, 1u);
    if (p < SEL_CAP) { sv[p] = cv[i]; si[p] = ci[i]; }
  } else if (bits == T) {
    unsigned e = atomicAdd(eqCnt, 1u);
    if (e < eqTake) {
      unsigned p = atomicAdd(selCnt, 1u);
      if (p < SEL_CAP) { sv[p] = cv[i]; si[p] = ci[i]; }
    }
  }
}

// One-workgroup bitonic sort of 4096 (val,info) pairs, descending (32KB LDS).
__global__ __launch_bounds__(512) void k_sort(float* __restrict__ sv,
                                              unsigned* __restrict__ si) {
  __shared__ float v[SEL_CAP];
  __shared__ unsigned f[SEL_CAP];
  for (int i = threadIdx.x; i < SEL_CAP; i += 512) { v[i] = sv[i]; f[i] = si[i]; }
  __syncthreads();
  for (int k = 2; k <= SEL_CAP; k <<= 1)
    for (int j = k >> 1; j > 0; j >>= 1) {
      for (int i = threadIdx.x; i < SEL_CAP; i += 512) {
        int ixj = i ^ j;
        if (ixj > i) {
          bool desc = ((i & k) == 0);
          bool sw = desc ? (v[i] < v[ixj]) : (v[i] > v[ixj]);
          if (sw) {
            float tv = v[i]; v[i] = v[ixj]; v[ixj] = tv;
            unsigned tf = f[i]; f[i] = f[ixj]; f[ixj] = tf;
          }
        }
      }
      __syncthreads();
    }
  for (int i = threadIdx.x; i < SEL_CAP; i += 512) { sv[i] = v[i]; si[i] = f[i]; }
}

// ---------------------------------------------------------------------------
// LAFs + responses.  lafs_d = [[3*sig*pd, 0, col*pd],[0, 3*sig*pd, row*pd]]
// ---------------------------------------------------------------------------
__global__ void k_lafs(const float* __restrict__ sv, const unsigned* __restrict__ si,
                       float* __restrict__ out_lafs, float* __restrict__ out_resp,
                       float s1, float s2, float s3) {
  int i = blockIdx.x * blockDim.x + threadIdx.x;
  if (i >= N_FEATS) return;
  unsigned info = si[i];
  int oct = (info >> 24) & 3, lvl = (info >> 22) & 3;
  float row = (float)((info >> 11) & 0x7FF), col = (float)(info & 0x7FF);
  float pd = (float)(1 << oct);
  float sg = (lvl == 2) ? s2 : (lvl == 3) ? s3 : s1;
  float a = 3.0f * sg * pd;
  float* L = out_lafs + (size_t)i * 6;
  L[0] = a;    L[1] = 0.0f; L[2] = col * pd;
  L[3] = 0.0f; L[4] = a;    L[5] = row * pd;
  out_resp[i] = sv[i];
}

// ---------------------------------------------------------------------------
// 32x32 bilinear patches from the ORIGINAL image (L2-resident, 16MB)
// ---------------------------------------------------------------------------
__global__ __launch_bounds__(256) void k_patch(const float* __restrict__ img,
                                               const unsigned* __restrict__ si,
                                               float* __restrict__ patches,
                                               float s1, float s2, float s3,
                                               int H, int W) {
  int fidx = blockIdx.x;
  unsigned info = si[fidx];
  int oct = (info >> 24) & 3, lvl = (info >> 22) & 3;
  float row = (float)((info >> 11) & 0x7FF), col = (float)(info & 0x7FF);
  float pd = (float)(1 << oct);
  float sg = (lvl == 2) ? s2 : (lvl == 3) ? s3 : s1;
  float a = 3.0f * sg * pd;
  float t0 = col * pd, t1 = row * pd;
#pragma unroll
  for (int t = 0; t < 4; ++t) {
    int pix = threadIdx.x + 256 * t;
    int iy = pix >> 5, jx = pix & 31;
    float gu = -1.0f + 2.0f * (float)jx / 31.0f;
    float gv = -1.0f + 2.0f * (float)iy / 31.0f;
    float xs = fminf(fmaxf(t0 + a * gu, 0.0f), (float)W - 1.001f);
    float ys = fminf(fmaxf(t1 + a * gv, 0.0f), (float)H - 1.001f);
    int x0 = (int)floorf(xs), y0 = (int)floorf(ys);
    float wx = xs - (float)x0, wy = ys - (float)y0;
    const float* p0 = img + (size_t)y0 * W + x0;
    float Ia = p0[0], Ib = p0[1], Ic = p0[W], Id = p0[W + 1];
    patches[(size_t)fidx * 1024 + pix] =
        Ia * (1 - wx) * (1 - wy) + Ib * wx * (1 - wy)
      + Ic * (1 - wx) * wy + Id * wx * wy;
  }
}

// ---------------------------------------------------------------------------
// Host-side blur dispatch: radii {4,5,6,8}, dims {2048,1024,512} are static.
// ---------------------------------------------------------------------------
static void blur_pair(const float* src, float* tmpb, float* dst, int dim,
                      double s, hipStream_t stream) {
  const int r = std::max(1, (int)std::ceil(3.0 * s));
  const int blocks = (dim / 16) * (dim / 16) / WPB;
  const float sf = (float)s;
#define BLUR_CASE(RR, DD)                                                  \
    k_blur_h_t<RR, DD><<<blocks, 128, 0, stream>>>(src, tmpb, sf);         \
    k_blur_v_t<RR, DD><<<blocks, 128, 0, stream>>>(tmpb, dst, sf);
#define BLUR_SWITCH_R(DD)                                                  \
    switch (r) {                                                           \
      case 4: { BLUR_CASE(4, DD) } break;                                  \
      case 5: { BLUR_CASE(5, DD) } break;                                  \
      case 6: { BLUR_CASE(6, DD) } break;                                  \
      case 8: { BLUR_CASE(8, DD) } break;                                  \
      default:                                                             \
        k_blur_h_gen<<<blocks, 128, 0, stream>>>(src, tmpb, DD, sf, r);    \
        k_blur_v_gen<<<blocks, 128, 0, stream>>>(tmpb, dst, DD, DD, sf, r);\
    }
  switch (dim) {
    case 2048: BLUR_SWITCH_R(2048) break;
    case 1024: BLUR_SWITCH_R(1024) break;
    default:   BLUR_SWITCH_R(512)  break;
  }
#undef BLUR_SWITCH_R
#undef BLUR_CASE
}

// ---------------------------------------------------------------------------
extern "C" void kernel_launch(void* const* d_in, const int* in_sizes, int n_in,
                              void* d_out, int out_size, void* d_ws, size_t ws_size,
                              hipStream_t stream) {
  (void)in_sizes; (void)n_in; (void)out_size; (void)ws_size;
  const float* x = (const float*)d_in[0];
  float* out = (float*)d_out;
  const int H0 = 2048, W0 = 2048;
  const size_t S0 = (size_t)H0 * W0;

  // workspace layout (floats)
  float* base = (float*)d_ws;
  float* lvA  = base;
  float* lvB  = base + S0;
  float* tmp  = base + 2 * S0;
  float* rs[5];
  for (int i = 0; i < 5; ++i) rs[i] = base + (3 + i) * S0;
  float* down = base + 8 * S0;                   // S0/4
  float*    cv = base + 8 * S0 + S0 / 4;         // CAND_MAX floats
  unsigned* ci = (unsigned*)(cv + CAND_MAX);     // CAND_MAX uints
  float*    sv = (float*)(ci + CAND_MAX);        // SEL_CAP
  unsigned* si = (unsigned*)(sv + SEL_CAP);      // SEL_CAP
  unsigned* cnt    = si + SEL_CAP;               // [0]=cand [1]=sel [2]=eq
  unsigned* selCnt = cnt + 1;
  unsigned* eqCnt  = cnt + 2;
  unsigned* state  = cnt + 4;                    // 4 words
  unsigned* hist   = cnt + 8;                    // 256 words

  double sig[5];
  for (int i = 0; i < 5; ++i) sig[i] = 1.6 * std::pow(2.0, (double)i / 3.0);
  const float s1 = (float)sig[1], s2 = (float)sig[2], s3 = (float)sig[3];

  hipMemsetAsync(cnt, 0, 3 * sizeof(unsigned), stream);
  hipMemsetAsync(sv, 0, SEL_CAP * sizeof(float), stream);
  hipMemsetAsync(si, 0, SEL_CAP * sizeof(unsigned), stream);

  // initial blur: sqrt(max(1.6^2 - 0.25, 0.01))
  blur_pair(x, tmp, lvA, W0, std::sqrt(std::max(1.6 * 1.6 - 0.25, 0.01)), stream);

  for (int o = 0; o < 3; ++o) {
    const int H = H0 >> o, W = W0 >> o;
    int lw = 0; while ((1 << lw) < W) ++lw;       // log2(W)
    const int npix = H * W, eb = (npix + 255) / 256;
    float* cur = (o == 0) ? lvA : down;           // level 0 of this octave
    k_hessian<<<eb, 256, 0, stream>>>(cur, rs[0], H, lw,
                                      (float)std::pow(sig[0], 4.0));
    const float* prev = cur;
    float* lev3 = nullptr;
    for (int i = 1; i <= 4; ++i) {
      float* dst = (i & 1) ? lvB : lvA;           // ping-pong B,A,B,A
      double ds = std::sqrt(sig[i] * sig[i] - sig[i - 1] * sig[i - 1]);
      blur_pair(prev, tmp, dst, W, ds, stream);
      k_hessian<<<eb, 256, 0, stream>>>(dst, rs[i], H, lw,
                                        (float)std::pow(sig[i], 4.0));
      if (i == 3) lev3 = dst;
      prev = dst;
    }
    const int b = std::max(1, 16 >> o);
    for (int li = 1; li <= 3; ++li)
      k_nms<<<eb, 256, 0, stream>>>(rs[li - 1], rs[li], rs[li + 1], H, lw, b,
                                    o, li, cv, ci, cnt, CAND_MAX);
    if (o < 2) {
      int H2 = H / 2, W2 = W / 2, lw2 = lw - 1;
      k_downsample<<<(H2 * W2 + 255) / 256, 256, 0, stream>>>(lev3, down,
                                                              H2 * W2, lw2);
    }
  }

  // global exact top-4000 (equivalent to per-level-then-global top_k)
  k_sel_init<<<1, 1, 0, stream>>>(state);
  const int hb = CAND_MAX / 256;
  for (int rnd = 0; rnd < 4; ++rnd) {
    hipMemsetAsync(hist, 0, 256 * sizeof(unsigned), stream);
    k_hist<<<hb, 256, 0, stream>>>(cv, cnt, hist, state, rnd, CAND_MAX);
    k_scan<<<1, 1, 0, stream>>>(hist, state, rnd);
  }
  k_compact<<<hb, 256, 0, stream>>>(cv, ci, cnt, state, sv, si, selCnt, eqCnt,
                                    CAND_MAX);
  k_sort<<<1, 512, 0, stream>>>(sv, si);

  // outputs: lafs_d [4000*6] | patches [4000*1024] | top_resp [4000]
  float* out_lafs = out;
  float* out_patch = out + (size_t)N_FEATS * 6;
  float* out_resp = out + (size_t)N_FEATS * 6 + (size_t)N_FEATS * 1024;
  k_lafs<<<(N_FEATS + 255) / 256, 256, 0, stream>>>(sv, si, out_lafs, out_resp,
                                                    s1, s2, s3);
  k_patch<<<N_FEATS, 256, 0, stream>>>(x, si, out_patch, s1, s2, s3, H0, W0);
}